// PointConv_29798483100371
// MI455X (gfx1250) — compile-verified
//
#include <hip/hip_runtime.h>
#include <stdint.h>

typedef float v2f __attribute__((ext_vector_type(2)));
typedef float v8f __attribute__((ext_vector_type(8)));

#define B_    4
#define K_    2048
#define P_    4096
#define NB_   16
#define CIN_  64
#define CMID_ 32
#define FH_   256
#define COUT_ 128

// ---------------------------------------------------------------------------
// Kernel 1: brute-force KNN (top-16 smallest squared distances per key).
// One wave32 per key; per-lane register sorted list, butterfly-shuffle merge.
// ---------------------------------------------------------------------------
__global__ __launch_bounds__(256) void knn_kernel(
    const float* __restrict__ keys, const float* __restrict__ points,
    int* __restrict__ nbrIdx, float* __restrict__ nbrRel)
{
    __shared__ float sd[8][32 * NB_];
    __shared__ int   si[8][32 * NB_];
    __shared__ int   win[8][NB_];

    const int wave = threadIdx.x >> 5;
    const int lane = threadIdx.x & 31;
    const int kk   = blockIdx.x * 8 + wave;      // global key row in [0, B*K)
    const int b    = kk >> 11;                   // / K_
    const float kx = keys[kk * 3 + 0];
    const float ky = keys[kk * 3 + 1];
    const float kz = keys[kk * 3 + 2];
    const float* pts = points + (size_t)b * P_ * 3;

    float bd[NB_]; int bi[NB_];
#pragma unroll
    for (int j = 0; j < NB_; ++j) { bd[j] = 3.0e38f; bi[j] = 0; }

    // each lane scans P/32 = 128 points, keeps sorted top-16 in registers
    for (int p = lane; p < P_; p += 32) {
        const float dx = pts[p * 3 + 0] - kx;
        const float dy = pts[p * 3 + 1] - ky;
        const float dz = pts[p * 3 + 2] - kz;
        const float d  = dx * dx + dy * dy + dz * dz;
        if (d < bd[NB_ - 1]) {
            bd[NB_ - 1] = d; bi[NB_ - 1] = p;
#pragma unroll
            for (int j = NB_ - 1; j > 0; --j) {
                if (bd[j] < bd[j - 1]) {
                    const float td = bd[j]; bd[j] = bd[j - 1]; bd[j - 1] = td;
                    const int   ti = bi[j]; bi[j] = bi[j - 1]; bi[j - 1] = ti;
                }
            }
        }
    }
#pragma unroll
    for (int j = 0; j < NB_; ++j) {
        sd[wave][lane * NB_ + j] = bd[j];
        si[wave][lane * NB_ + j] = bi[j];
    }
    __syncthreads();

    // merge 32 sorted lists: 16 rounds of wave-wide lexicographic min
    int head = 0;
    for (int r = 0; r < NB_; ++r) {
        float d  = (head < NB_) ? sd[wave][lane * NB_ + head] : 3.2e38f;
        int   id = (head < NB_) ? si[wave][lane * NB_ + head] : 0x7fffffff;
        int   ln = lane;
#pragma unroll
        for (int off = 16; off >= 1; off >>= 1) {
            const float od  = __shfl_xor(d, off, 32);
            const int   oid = __shfl_xor(id, off, 32);
            const int   oln = __shfl_xor(ln, off, 32);
            if (od < d || (od == d && oid < id)) { d = od; id = oid; ln = oln; }
        }
        if (lane == 0) win[wave][r] = id;
        if (lane == ln) head++;
    }
    __syncthreads();

    if (lane < NB_) {
        const int id = win[wave][lane];
        nbrIdx[kk * NB_ + lane] = id;
        nbrRel[(kk * NB_ + lane) * 3 + 0] = pts[id * 3 + 0] - kx;
        nbrRel[(kk * NB_ + lane) * 3 + 1] = pts[id * 3 + 1] - ky;
        nbrRel[(kk * NB_ + lane) * 3 + 2] = pts[id * 3 + 2] - kz;
    }
}

// ---------------------------------------------------------------------------
// Kernel 2: fused edge-MLP + aggregation + fc0 (WMMA) + fc1 (WMMA).
// 16 keys per 256-thread block; e[16][2048] tile never leaves LDS.
// Neighbor features gathered memory->LDS via the CDNA5 async copy engine.
// ---------------------------------------------------------------------------
__global__ __launch_bounds__(256) void pconv_kernel(
    const float* __restrict__ feats,
    const float* __restrict__ wc0W, const float* __restrict__ wc0b,
    const float* __restrict__ wc1W, const float* __restrict__ wc1b,
    const float* __restrict__ wc2W, const float* __restrict__ wc2b,
    const float* __restrict__ fc0W, const float* __restrict__ fc0b,
    const float* __restrict__ fc1W, const float* __restrict__ fc1b,
    const int*   __restrict__ nbrIdx, const float* __restrict__ nbrRel,
    float* __restrict__ out)
{
    extern __shared__ float smem[];
    const int EROW = 2052;   // 2048 + 4 pad: lanes 0..15 hit distinct LDS banks
    const int HROW = 260;    // 256 + 4 pad
    float* eL = smem;                       // [16][EROW]
    float* hL = eL + 16 * EROW;             // [16][HROW]
    float* mB = hL + 16 * HROW;             // [16][16][32]  edge-MLP outputs
    float* fB = mB + 16 * 16 * 32;          // [16][16][64]  gathered feats
    float* wB = fB + 16 * 16 * 64;          // 2240 staged MLP weights

    const int tid = threadIdx.x;
    const int kk0 = blockIdx.x * 16;        // first global key row of tile
    const int b   = kk0 >> 11;

    // stage the tiny weight-MLP parameters into LDS (broadcast reads later)
    for (int i = tid; i < 2240; i += 256) {
        float v;
        if      (i <   96) v = wc0W[i];
        else if (i <  128) v = wc0b[i - 96];
        else if (i < 1152) v = wc1W[i - 128];
        else if (i < 1184) v = wc1b[i - 1152];
        else if (i < 2208) v = wc2W[i - 1184];
        else               v = wc2b[i - 2208];
        wB[i] = v;
    }
    __syncthreads();
    const float* Wc0 = wB;          // [3][32]
    const float* bc0 = wB + 96;
    const float* Wc1 = wB + 128;    // [32][32]
    const float* bc1 = wB + 1152;
    const float* Wc2 = wB + 1184;   // [32][32]
    const float* bc2 = wB + 2208;

    // ---- Phase A: one thread per (key, neighbor) edge: 3->32->32->32 MLP ----
    {
        const int key = tid >> 4, n = tid & 15;
        const int kk  = kk0 + key;

        // kick off the async memory->LDS gather of this edge's 64-ch feature
        // row FIRST, so the copy engine overlaps the MLP math below.
        // GLOBAL_LOAD_ASYNC_TO_LDS_B128: vdst = LDS byte address (low 32 bits
        // of the generic shared pointer), vaddr = 64-bit global address.
        {
            const int idx = nbrIdx[kk * NB_ + n];
            const float* fsrc = feats + ((size_t)b * P_ + idx) * CIN_;
            float* fdst = fB + (key * 16 + n) * CIN_;
#pragma unroll
            for (int c = 0; c < CIN_; c += 4) {
                const uint32_t ldsAddr = (uint32_t)(uintptr_t)(fdst + c);
                asm volatile("global_load_async_to_lds_b128 %0, %1, off"
                             :: "v"(ldsAddr), "v"(fsrc + c)
                             : "memory");
            }
        }

        const float r0 = nbrRel[(kk * NB_ + n) * 3 + 0];
        const float r1 = nbrRel[(kk * NB_ + n) * 3 + 1];
        const float r2 = nbrRel[(kk * NB_ + n) * 3 + 2];

        float h0[32];
#pragma unroll
        for (int j = 0; j < 32; ++j) {
            const float s = r0 * Wc0[j] + r1 * Wc0[32 + j] + r2 * Wc0[64 + j] + bc0[j];
            h0[j] = s > 0.f ? s : 0.f;
        }
        float h1[32];
#pragma unroll
        for (int j = 0; j < 32; ++j) {
            float s = bc1[j];
#pragma unroll
            for (int i = 0; i < 32; ++i) s += h0[i] * Wc1[i * 32 + j];
            h1[j] = s > 0.f ? s : 0.f;
        }
        float* mrow = mB + (key * 16 + n) * 32;
#pragma unroll
        for (int j = 0; j < 32; ++j) {
            float s = bc2[j];
#pragma unroll
            for (int i = 0; i < 32; ++i) s += h1[i] * Wc2[i * 32 + j];
            mrow[j] = s;
        }
    }
    // drain this wave's async copies before the block-wide barrier
#if __has_builtin(__builtin_amdgcn_s_wait_asynccnt)
    __builtin_amdgcn_s_wait_asynccnt(0);
#else
    asm volatile("s_wait_asynccnt 0" ::: "memory");
#endif
    __syncthreads();

    // ---- Phase A2: e[key][mi*64+c] = sum_n m[key][n][mi] * f[key][n][c] ----
    for (int t = tid; t < 512; t += 256) {      // 16 keys x 32 mi
        const int key = t >> 5, mi = t & 31;
#pragma unroll
        for (int cb = 0; cb < 4; ++cb) {
            float acc[16];
#pragma unroll
            for (int c = 0; c < 16; ++c) acc[c] = 0.f;
#pragma unroll
            for (int n = 0; n < 16; ++n) {
                const float w  = mB[(key * 16 + n) * 32 + mi];
                const float* fr = fB + (key * 16 + n) * CIN_ + cb * 16;
#pragma unroll
                for (int c = 0; c < 16; ++c) acc[c] += w * fr[c];
            }
            float* er = eL + key * EROW + mi * 64 + cb * 16;
#pragma unroll
            for (int c = 0; c < 16; ++c) er[c] = acc[c];
        }
    }
    __syncthreads();

    const int lane = tid & 31;
    const int wave = tid >> 5;
    const int mRow = lane & 15;             // A: M index, D: N index held by this lane
    const int kOff = (lane >> 4) << 1;      // A: lanes>=16 carry K=+2,+3
    const int mBase = (lane >> 4) * 8;      // D: lanes>=16 hold M=8..15

    // ---- Phase B: h = relu(e @ fc0W + fc0b), M=16 K=2048 N=256, WMMA f32 ----
    {
        const int c0 = wave * 32 + (lane & 15);     // this wave: cols [w*32, w*32+32)
        v8f acc0 = {0.f,0.f,0.f,0.f,0.f,0.f,0.f,0.f};
        v8f acc1 = {0.f,0.f,0.f,0.f,0.f,0.f,0.f,0.f};
        const float* eRow = eL + mRow * EROW;
        for (int kb = 0; kb < 2048; kb += 4) {
            v2f a;
            a.x = eRow[kb + kOff];
            a.y = eRow[kb + kOff + 1];
            const float* wp = fc0W + (size_t)(kb + kOff) * FH_;
            __builtin_prefetch(wp + 4 * FH_, 0, 1);   // stream next fc0W rows (L2-resident)
            v2f b0; b0.x = wp[c0];       b0.y = wp[FH_ + c0];
            v2f b1; b1.x = wp[c0 + 16];  b1.y = wp[FH_ + c0 + 16];
            acc0 = __builtin_amdgcn_wmma_f32_16x16x4_f32(false, a, false, b0,
                                                         (short)0, acc0, false, false);
            acc1 = __builtin_amdgcn_wmma_f32_16x16x4_f32(false, a, false, b1,
                                                         (short)0, acc1, false, false);
        }
#pragma unroll
        for (int v = 0; v < 8; ++v) {
            const int row = mBase + v;
            const float x0 = acc0[v] + fc0b[c0];
            const float x1 = acc1[v] + fc0b[c0 + 16];
            hL[row * HROW + c0]      = x0 > 0.f ? x0 : 0.f;
            hL[row * HROW + c0 + 16] = x1 > 0.f ? x1 : 0.f;
        }
    }
    __syncthreads();

    // ---- Phase C: out = h @ fc1W + fc1b, M=16 K=256 N=128, WMMA f32 ----
    {
        const int col = wave * 16 + (lane & 15);    // 8 waves x 16 cols = 128
        v8f acc = {0.f,0.f,0.f,0.f,0.f,0.f,0.f,0.f};
        const float* hRow = hL + mRow * HROW;
        for (int kb = 0; kb < 256; kb += 4) {
            v2f a;
            a.x = hRow[kb + kOff];
            a.y = hRow[kb + kOff + 1];
            const float* wp = fc1W + (size_t)(kb + kOff) * COUT_;
            v2f bv; bv.x = wp[col]; bv.y = wp[COUT_ + col];
            acc = __builtin_amdgcn_wmma_f32_16x16x4_f32(false, a, false, bv,
                                                        (short)0, acc, false, false);
        }
        const float bias = fc1b[col];
#pragma unroll
        for (int v = 0; v < 8; ++v) {
            const int row = mBase + v;
            out[(size_t)(kk0 + row) * COUT_ + col] = acc[v] + bias;
        }
    }
}

// ---------------------------------------------------------------------------
extern "C" void kernel_launch(void* const* d_in, const int* in_sizes, int n_in,
                              void* d_out, int out_size, void* d_ws, size_t ws_size,
                              hipStream_t stream)
{
    const float* keys   = (const float*)d_in[0];
    const float* points = (const float*)d_in[1];
    const float* feats  = (const float*)d_in[2];
    const float* wc0W   = (const float*)d_in[3];
    const float* wc0b   = (const float*)d_in[4];
    const float* wc1W   = (const float*)d_in[5];
    const float* wc1b   = (const float*)d_in[6];
    const float* wc2W   = (const float*)d_in[7];
    const float* wc2b   = (const float*)d_in[8];
    const float* fc0W   = (const float*)d_in[9];
    const float* fc0b   = (const float*)d_in[10];
    const float* fc1W   = (const float*)d_in[11];
    const float* fc1b   = (const float*)d_in[12];
    float* out = (float*)d_out;

    const int rows = B_ * K_;                       // 8192 key rows
    int*   nbrIdx = (int*)d_ws;                     // [8192][16]
    float* nbrRel = (float*)((char*)d_ws + (size_t)rows * NB_ * sizeof(int));

    knn_kernel<<<rows / 8, 256, 0, stream>>>(keys, points, nbrIdx, nbrRel);

    const size_t shmem = (size_t)(16 * 2052 + 16 * 260 + 16 * 16 * 32 +
                                  16 * 16 * 64 + 2240) * sizeof(float);  // ~249 KB
    pconv_kernel<<<rows / 16, 256, shmem, stream>>>(
        feats, wc0W, wc0b, wc1W, wc1b, wc2W, wc2b,
        fc0W, fc0b, fc1W, fc1b, nbrIdx, nbrRel, out);
}